// QFD2Loss_79130477461937
// MI455X (gfx1250) — compile-verified
//
#include <hip/hip_runtime.h>

typedef float v2f __attribute__((ext_vector_type(2)));
typedef float v8f __attribute__((ext_vector_type(8)));

#define LDIM 64

// Upper-triangle tile pairs (tj <= tk) of the 4x4 tile grid of G (64x64).
__device__ __constant__ static const int kTJ[10] = {0,0,0,0,1,1,1,2,2,3};
__device__ __constant__ static const int kTK[10] = {0,1,2,3,1,2,3,2,3,3};

// Stage 1: per-wave partial Gram accumulation via V_WMMA_F32_16X16X4_F32,
// folded immediately against analytic A[j,k] = 1 - |j-k|/63.
__global__ __launch_bounds__(256) void qfd2_gram_kernel(
    const float* __restrict__ D,
    const float* __restrict__ P,
    float* __restrict__ partials,
    int B)
{
    const int lane = threadIdx.x & 31;
    const int wid  = threadIdx.x >> 5;
    const int wpb  = blockDim.x >> 5;
    const int gwave  = blockIdx.x * wpb + wid;
    const int nwaves = gridDim.x * wpb;

    const int lo = lane & 15;   // column-in-tile / N index
    const int hi = lane >> 4;   // half-wave selector

    v8f acc[10];
    #pragma unroll
    for (int p = 0; p < 10; ++p) { v8f z = {}; acc[p] = z; }

    const int tot_ksteps = B >> 2;   // 4 batch rows per WMMA k-step
    for (int ks = gwave; ks < tot_ksteps; ks += nwaves) {
        const int r0 = (ks << 2) + (hi << 1);  // first of 2 rows this lane reads
        const float* dRow = D + (size_t)r0 * LDIM + lo;
        const float* pRow = P + (size_t)r0 * LDIM + lo;

        // q[t] = A-operand registers for column tile t: Q[r0+v, 16t+lo]
        v2f q[4];
        #pragma unroll
        for (int t = 0; t < 4; ++t) {
            q[t].x = dRow[t * 16]        - pRow[t * 16];
            q[t].y = dRow[LDIM + t * 16] - pRow[LDIM + t * 16];
        }

        #pragma unroll
        for (int p = 0; p < 10; ++p) {
            acc[p] = __builtin_amdgcn_wmma_f32_16x16x4_f32(
                false, q[kTJ[p]], false, q[kTK[p]],
                (short)0, acc[p], false, false);
        }
    }

    // Epilogue: s = sum_{j,k} A[j,k] * G_partial[j,k], using symmetry (x2 off-diag tiles)
    float s = 0.0f;
    #pragma unroll
    for (int p = 0; p < 10; ++p) {
        const int jb = kTJ[p] * 16;
        const int kb = kTK[p] * 16;
        const float mult = (kTJ[p] == kTK[p]) ? 1.0f : 2.0f;
        #pragma unroll
        for (int i = 0; i < 8; ++i) {
            const int jj = jb + i + (hi << 3);   // D-layout: M = i (+8 for lanes 16-31)
            const int kk = kb + lo;              // D-layout: N = lane & 15
            int d = jj - kk; d = (d < 0) ? -d : d;
            const float w = 1.0f - (float)d * (1.0f / 63.0f);
            s += (mult * w) * acc[p][i];
        }
    }

    // Wave32 reduction
    #pragma unroll
    for (int off = 16; off > 0; off >>= 1)
        s += __shfl_xor(s, off, 32);

    __shared__ float red[32];
    if (lane == 0) red[wid] = s;
    __syncthreads();
    if (threadIdx.x == 0) {
        float t = 0.0f;
        for (int i = 0; i < wpb; ++i) t += red[i];
        partials[blockIdx.x] = t;
    }
}

// Stage 2: reduce block partials, scale by 1/B.
__global__ __launch_bounds__(256) void qfd2_reduce_kernel(
    const float* __restrict__ partials, int n,
    float* __restrict__ out, float invB)
{
    float s = 0.0f;
    for (int i = threadIdx.x; i < n; i += blockDim.x) s += partials[i];
    #pragma unroll
    for (int off = 16; off > 0; off >>= 1)
        s += __shfl_xor(s, off, 32);

    __shared__ float red[8];
    const int lane = threadIdx.x & 31;
    const int wid  = threadIdx.x >> 5;
    if (lane == 0) red[wid] = s;
    __syncthreads();
    if (threadIdx.x == 0) {
        float t = 0.0f;
        for (int i = 0; i < (int)(blockDim.x >> 5); ++i) t += red[i];
        out[0] = t * invB;
    }
}

extern "C" void kernel_launch(void* const* d_in, const int* in_sizes, int n_in,
                              void* d_out, int out_size, void* d_ws, size_t ws_size,
                              hipStream_t stream) {
    const float* D = (const float*)d_in[0];
    const float* P = (const float*)d_in[1];
    float* out = (float*)d_out;
    float* ws  = (float*)d_ws;

    const int B = in_sizes[0] / LDIM;   // 262144
    const int blocks  = 256;            // 2048 waves; 128 rows per wave
    const int threads = 256;            // 8 waves/block

    qfd2_gram_kernel<<<blocks, threads, 0, stream>>>(D, P, ws, B);
    qfd2_reduce_kernel<<<1, 256, 0, stream>>>(ws, blocks, out, 1.0f / (float)B);
}